// Model_87290915324105
// MI455X (gfx1250) — compile-verified
//
#include <hip/hip_runtime.h>

// Problem constants (from reference)
#define USERS   100000
#define ITEMS   100000
#define NNODES  (USERS + ITEMS)     // 200000
#define LATDIM  64                  // 16 float4 per node row
#define NEDGES  1250000
#define LEAKY   0.5f

typedef float v4f __attribute__((ext_vector_type(4)));

// ---------------------------------------------------------------------------
// init: cur = x, acc(out) = x, agg = 0     (x = concat(u_embeds, i_embeds))
// index space: NNODES*16 v4f elements = 3,200,000 (exactly 12500 blocks * 256)
// Embeddings are read exactly once per call -> non-temporal loads so they
// don't displace the L2-resident cur/agg/acc working set.
// ---------------------------------------------------------------------------
__global__ void gnn_init(const v4f* __restrict__ u4,
                         const v4f* __restrict__ i4,
                         v4f* __restrict__ cur4,
                         v4f* __restrict__ agg4,
                         v4f* __restrict__ out4) {
    int i = blockIdx.x * 256 + threadIdx.x;           // < NNODES*16
    const v4f* src = (i < USERS * 16) ? (u4 + i) : (i4 + (i - USERS * 16));
    v4f x = __builtin_nontemporal_load(src);          // single b128 load
    cur4[i] = x;                                      // RT: hot in L2
    out4[i] = x;                                      // RT: re-read in post
    agg4[i] = (v4f)0.f;
}

// ---------------------------------------------------------------------------
// scatter: for each edge e: agg[row[e]] += cur[col[e]] * val[e]
//
// Layout: 16 lanes per edge (one float4 = 16B of the 256B row per lane),
// 2 edges per wave32, 16 edges per 256-thread block. NEDGES % 16 == 0 ->
// exact grid (78125 blocks), EXEC all-ones at every async op.
//
// CDNA5 path: gathered row DMA'd global->LDS with
// GLOBAL_LOAD_ASYNC_TO_LDS_B128 (ASYNCcnt), s_wait_asynccnt, ds_load_b128
// read-back, scale, then 4 native f32 device-scope atomics into L2 sharing
// one address VGPR pair via offset: immediates.
// Edge metadata is streamed with NT hints (read once per layer).
// ---------------------------------------------------------------------------
__global__ void gnn_scatter(const long long* __restrict__ rows,
                            const long long* __restrict__ cols,
                            const float* __restrict__ vals,
                            const float* __restrict__ cur,
                            float* __restrict__ agg) {
    __shared__ v4f stage[16][16];                     // 16 edges x 256B = 4KB

    int t   = threadIdx.x;                            // 0..255
    int sub = t >> 4;                                 // edge slot in block
    int dim = t & 15;                                 // float4 index in row
    long long e = (long long)blockIdx.x * 16 + sub;   // always < NEDGES

    int   r = (int)__builtin_nontemporal_load(rows + e);
    int   c = (int)__builtin_nontemporal_load(cols + e);
    float v = __builtin_nontemporal_load(vals + e);

    // global source address for this lane's 16B of cur[c]
    const float* src = cur + (size_t)c * LATDIM + (dim << 2);
    // LDS byte offset of this lane's staging slot (addrspace(3) offset is the
    // low 32 bits of the flat shared-aperture address)
    unsigned ldsoff = (unsigned)(size_t)(void*)&stage[sub][dim];

    // async DMA 16B global -> LDS (tracked by ASYNCcnt)
    asm volatile("global_load_async_to_lds_b128 %0, %1, off"
                 :: "v"(ldsoff), "v"(src)
                 : "memory");
    // wait for our own async LDS writes to land
    asm volatile("s_wait_asynccnt 0" ::: "memory");

    v4f m = stage[sub][dim];
    m.x *= v; m.y *= v; m.z *= v; m.w *= v;

    float* dst = agg + (size_t)r * LATDIM + (dim << 2);
    // 4 native f32 atomics, device scope (cross-WGP reduction), RMW in L2
    asm volatile(
        "global_atomic_add_f32 %0, %1, off scope:SCOPE_DEV\n\t"
        "global_atomic_add_f32 %0, %2, off offset:4 scope:SCOPE_DEV\n\t"
        "global_atomic_add_f32 %0, %3, off offset:8 scope:SCOPE_DEV\n\t"
        "global_atomic_add_f32 %0, %4, off offset:12 scope:SCOPE_DEV"
        :: "v"(dst), "v"(m.x), "v"(m.y), "v"(m.z), "v"(m.w)
        : "memory");
}

// ---------------------------------------------------------------------------
// post: cur = leaky_relu(agg, 0.5); acc(out) += cur; agg = 0 (for next layer)
// ---------------------------------------------------------------------------
__global__ void gnn_post(v4f* __restrict__ agg4,
                         v4f* __restrict__ cur4,
                         v4f* __restrict__ out4) {
    int i = blockIdx.x * 256 + threadIdx.x;           // < NNODES*16
    v4f a = agg4[i];
    v4f rr;
    rr.x = a.x > 0.f ? a.x : LEAKY * a.x;
    rr.y = a.y > 0.f ? a.y : LEAKY * a.y;
    rr.z = a.z > 0.f ? a.z : LEAKY * a.z;
    rr.w = a.w > 0.f ? a.w : LEAKY * a.w;
    cur4[i] = rr;
    v4f o = out4[i];
    o.x += rr.x; o.y += rr.y; o.z += rr.z; o.w += rr.w;
    out4[i] = o;
    agg4[i] = (v4f)0.f;
}

// ---------------------------------------------------------------------------
extern "C" void kernel_launch(void* const* d_in, const int* in_sizes, int n_in,
                              void* d_out, int out_size, void* d_ws, size_t ws_size,
                              hipStream_t stream) {
    // inputs (reference dtypes): edge_index int64 [2,E], vals f32 [E],
    // u_embeds f32 [USERS,64], i_embeds f32 [ITEMS,64]
    const long long* rows = (const long long*)d_in[0];
    const long long* cols = rows + NEDGES;
    const float*     vals = (const float*)d_in[1];
    const float*     u    = (const float*)d_in[2];
    const float*     ie   = (const float*)d_in[3];
    float*           out  = (float*)d_out;            // acc [NNODES,64]

    // workspace: cur (51.2 MB) + agg (51.2 MB)
    float* cur = (float*)d_ws;
    float* agg = cur + (size_t)NNODES * LATDIM;

    const int elem_blocks = (NNODES * 16) / 256;      // 12500, exact
    const int edge_blocks = NEDGES / 16;              // 78125, exact

    gnn_init<<<elem_blocks, 256, 0, stream>>>(
        (const v4f*)u, (const v4f*)ie,
        (v4f*)cur, (v4f*)agg, (v4f*)out);

    for (int layer = 0; layer < 2; ++layer) {
        gnn_scatter<<<edge_blocks, 256, 0, stream>>>(rows, cols, vals, cur, agg);
        gnn_post<<<elem_blocks, 256, 0, stream>>>(
            (v4f*)agg, (v4f*)cur, (v4f*)out);
    }
}